// AttentionLayers_7559142441437
// MI455X (gfx1250) — compile-verified
//
#include <hip/hip_runtime.h>

typedef float v2f __attribute__((ext_vector_type(2)));
typedef float v8f __attribute__((ext_vector_type(8)));

#define BATCH 32
#define SLEN  2048
#define DDIM  1024
#define HEADS 4

// One block = one (batch, 64-wide d-stripe). 1024 threads = 32 waves.
// Wave w: dtile = w & 3 (16 cols each), schunk = w >> 2 (256 rows each).
// Each wave folds its 256x16 sub-panel into a 16-col sum using 64
// V_WMMA_F32_16X16X4_F32 ops with an all-ones A matrix (ones^T * X).
__global__ __launch_bounds__(1024)
void AttentionLayers_meanpool_wmma(const float* __restrict__ x,
                                   const int* __restrict__ lens,
                                   float* __restrict__ out) {
    __shared__ float partial[4][8][16];

    const int b      = blockIdx.x >> 4;        // 16 d-groups per batch
    const int dgroup = blockIdx.x & 15;
    const int d0     = dgroup * 64;            // 256B-aligned column window
    const int wave   = threadIdx.x >> 5;
    const int lane   = threadIdx.x & 31;
    const int dtile  = wave & 3;               // which 16-col sub-tile
    const int schunk = wave >> 2;              // which 256-row chunk
    const int col    = lane & 15;              // N = lane & 15 (both halves)
    const int k0     = (lane >> 4) << 1;       // K rows 0/1 (lanes 0-15), 2/3 (16-31)

    // Per-lane streaming pointer: column (d0 + dtile*16 + col), row (schunk*256 + k0)
    const float* p = x + (size_t)b * SLEN * DDIM
                       + (size_t)(schunk * 256 + k0) * DDIM
                       + (d0 + dtile * 16 + col);

    v2f a; a.x = 1.0f; a.y = 1.0f;             // all-ones 16x4 A matrix
    v8f c = {};                                 // fp32 accumulator (16x16, rows identical)

#pragma unroll 4
    for (int it = 0; it < 64; ++it) {
        v2f bv;
        bv.x = p[0];                            // row s+k0   , 64B fully-used segment/half-wave
        bv.y = p[DDIM];                         // row s+k0+1
        // D = ones(16x4) * B(4x16) + C  =>  every row of D = running column sums
        c = __builtin_amdgcn_wmma_f32_16x16x4_f32(
                /*neg_a=*/false, a, /*neg_b=*/false, bv,
                /*c_mod=*/(short)0, c, /*reuse_a=*/false, /*reuse_b=*/false);
        p += 4 * DDIM;                          // advance 4 rows
    }

    // Lane L, D-VGPR0 holds colsum for N = L & 15 (all D rows identical).
    if (lane < 16) partial[dtile][schunk][col] = c[0];
    __syncthreads();

    // 64 threads finish: combine 8 S-chunk partials, scale by 1/len, replicate x4 heads.
    if (threadIdx.x < 64) {
        const int t  = threadIdx.x;
        const int dt = t >> 4;
        const int cc = t & 15;
        float s = 0.0f;
#pragma unroll
        for (int w = 0; w < 8; ++w) s += partial[dt][w][cc];
        const float m = s * (1.0f / (float)lens[b]);
        // out[b, d*4 + h] = m for h=0..3  ->  one coalesced float4 per mean element
        ((float4*)out)[(size_t)b * (DDIM) + (d0 + t)] = make_float4(m, m, m, m);
    }
}

extern "C" void kernel_launch(void* const* d_in, const int* in_sizes, int n_in,
                              void* d_out, int out_size, void* d_ws, size_t ws_size,
                              hipStream_t stream) {
    (void)in_sizes; (void)n_in; (void)d_ws; (void)ws_size; (void)out_size;
    const float* x    = (const float*)d_in[0];   // [32, 2048, 1024] fp32
    const int*   lens = (const int*)d_in[1];     // [32] int32
    float*       out  = (float*)d_out;           // [32, 4096] fp32

    dim3 grid(BATCH * (DDIM / 64));              // 512 blocks
    dim3 block(1024);                            // 32 waves (wave32)
    AttentionLayers_meanpool_wmma<<<grid, block, 0, stream>>>(x, lens, out);
}